// EnergyGNN_GIN_22883585753797
// MI455X (gfx1250) — compile-verified
//
#include <hip/hip_runtime.h>

#define N_NODES   100000
#define N_EDGES   1600000
#define HID       64
#define NUM_GRAPHS 128
#define BN_EPS    1e-5f

typedef float v2f __attribute__((ext_vector_type(2)));
typedef float v8f __attribute__((ext_vector_type(8)));

// ---------------------------------------------------------------- zero fill
__global__ void k_zero(float* __restrict__ p, int n) {
    int i = blockIdx.x * blockDim.x + threadIdx.x;
    if (i < n) p[i] = 0.0f;
}

// ------------------------------------------------- edge scatter: agg[dst] += h[src]
// One thread per (edge, 4-float chunk). h is L2-resident (25.6MB << 192MB L2),
// so gathers + f32 atomics stay on-chip; only the edge stream hits HBM.
__global__ void k_scatter(const float* __restrict__ h,
                          const long long* __restrict__ ei,
                          float* agg) {
    long long t = (long long)blockIdx.x * blockDim.x + threadIdx.x;
    if (t >= (long long)N_EDGES * 16) return;
    int e = (int)(t >> 4);
    int c = ((int)t & 15) * 4;
    int s = (int)ei[e];
    int d = (int)ei[N_EDGES + e];
    const float4 v = *(const float4*)(h + (size_t)s * HID + c);
    float* p = agg + (size_t)d * HID + c;
    atomicAdd(p + 0, v.x);
    atomicAdd(p + 1, v.y);
    atomicAdd(p + 2, v.z);
    atomicAdd(p + 3, v.w);
}

// ------------------------------------------------------------- fused GIN MLP
// y = relu((x+agg) @ W1 + b1) @ W2 + b2, plus per-column sum / sum-of-squares
// for BatchNorm. One wave owns a 16-row x 64-col tile; f32 WMMA 16x16x4,
// K=64 in 16 steps, 4 N-tiles. Intermediate relu tile staged via wave-private
// LDS to re-layout D (C-layout) into A-layout for the second GEMM.
// NOTE: y may alias agg (rows are fully consumed before being overwritten).
__global__ void __launch_bounds__(128) k_gin_mlp(
        const float* __restrict__ xin, const float* agg,
        const float* __restrict__ W1, const float* __restrict__ B1,
        const float* __restrict__ W2, const float* __restrict__ B2,
        float* y, float* stats) {
    __shared__ float lds[4][16][68];   // +4 pad: conflict-free, 8B-aligned rows

    const int lane = threadIdx.x & 31;
    const int wv   = threadIdx.x >> 5;
    const int m0   = (blockIdx.x * 4 + wv) * 16;
    if (m0 >= N_NODES) return;         // wave-uniform: EXEC stays all-ones
    const int half = lane >> 4;        // lane half selects K pair (A) / K pair (B)
    const int lm   = lane & 15;        // A: row M; B/D: col N
    const size_t rowA = (size_t)(m0 + lm) * HID;

    // ---- GEMM1: (x + agg) @ W1
    v8f acc[4] = {};
    #pragma unroll
    for (int kk = 0; kk < 16; ++kk) {
        const int ka = kk * 4 + 2 * half;          // K base for this lane-half
        v2f a  = *(const v2f*)(xin + rowA + ka);
        v2f ag = *(const v2f*)(agg + rowA + ka);
        a.x += ag.x; a.y += ag.y;
        #pragma unroll
        for (int nt = 0; nt < 4; ++nt) {
            const int n = nt * 16 + lm;
            v2f b;
            b.x = W1[(ka + 0) * HID + n];
            b.y = W1[(ka + 1) * HID + n];
            acc[nt] = __builtin_amdgcn_wmma_f32_16x16x4_f32(
                false, a, false, b, (short)0, acc[nt], false, false);
        }
    }

    // ---- bias + relu, stash to LDS in row-major (wave-private, no barrier)
    #pragma unroll
    for (int nt = 0; nt < 4; ++nt) {
        const float bb = B1[nt * 16 + lm];
        #pragma unroll
        for (int j = 0; j < 8; ++j) {
            float v = fmaxf(acc[nt][j] + bb, 0.0f);
            lds[wv][j + 8 * half][nt * 16 + lm] = v;  // row = M, col = N
        }
    }

    // ---- GEMM2: relu(h1) @ W2
    v8f acc2[4] = {};
    #pragma unroll
    for (int kk = 0; kk < 16; ++kk) {
        const int ka = kk * 4 + 2 * half;
        v2f a = *(const v2f*)&lds[wv][lm][ka];
        #pragma unroll
        for (int nt = 0; nt < 4; ++nt) {
            const int n = nt * 16 + lm;
            v2f b;
            b.x = W2[(ka + 0) * HID + n];
            b.y = W2[(ka + 1) * HID + n];
            acc2[nt] = __builtin_amdgcn_wmma_f32_16x16x4_f32(
                false, a, false, b, (short)0, acc2[nt], false, false);
        }
    }

    // ---- +b2, store pre-BN y, accumulate BN column statistics
    #pragma unroll
    for (int nt = 0; nt < 4; ++nt) {
        const int n = nt * 16 + lm;
        const float bb = B2[n];
        float s = 0.0f, q = 0.0f;
        #pragma unroll
        for (int j = 0; j < 8; ++j) {
            float v = acc2[nt][j] + bb;
            y[(size_t)(m0 + j + 8 * half) * HID + n] = v;
            s += v;
            q += v * v;
        }
        s += __shfl_xor(s, 16);   // combine the two lane-halves (same column n)
        q += __shfl_xor(q, 16);
        if (lane < 16) {
            atomicAdd(&stats[n], s);
            atomicAdd(&stats[64 + n], q);
        }
    }
}

// ------------------------------------------------- BN finalize: scale/shift
__global__ void k_bn_final(const float* __restrict__ stats,
                           const float* __restrict__ gamma,
                           const float* __restrict__ beta,
                           float* __restrict__ bnp) {
    int i = threadIdx.x;
    if (i < HID) {
        const float invN = 1.0f / (float)N_NODES;
        float mean = stats[i] * invN;
        float var  = stats[64 + i] * invN - mean * mean;
        float sc   = gamma[i] * rsqrtf(var + BN_EPS);
        bnp[i]      = sc;
        bnp[64 + i] = beta[i] - mean * sc;
    }
}

// ------------------------------------------------- BN apply + relu
__global__ void k_bn_apply(const float* y, const float* __restrict__ bnp,
                           float* h) {
    long long t = (long long)blockIdx.x * blockDim.x + threadIdx.x;
    if (t >= (long long)N_NODES * 16) return;
    const int c = ((int)t & 15) * 4;
    const size_t base = (size_t)(t >> 4) * HID + c;
    float4 v = *(const float4*)(y + base);
    float4 r;
    r.x = fmaxf(fmaf(v.x, bnp[c + 0], bnp[64 + c + 0]), 0.0f);
    r.y = fmaxf(fmaf(v.y, bnp[c + 1], bnp[64 + c + 1]), 0.0f);
    r.z = fmaxf(fmaf(v.z, bnp[c + 2], bnp[64 + c + 2]), 0.0f);
    r.w = fmaxf(fmaf(v.w, bnp[c + 3], bnp[64 + c + 3]), 0.0f);
    *(float4*)(h + base) = r;
}

// ------------------------------------------------- per-graph sum pool
__global__ void k_pool(const float* __restrict__ h,
                       const long long* __restrict__ batch,
                       float* psum, float* pcnt) {
    long long t = (long long)blockIdx.x * blockDim.x + threadIdx.x;
    if (t >= (long long)N_NODES * 16) return;
    const int nd = (int)(t >> 4);
    const int c  = ((int)t & 15) * 4;
    const int g  = (int)batch[nd];
    const float4 v = *(const float4*)(h + (size_t)nd * HID + c);
    float* p = psum + (size_t)g * HID + c;
    atomicAdd(p + 0, v.x);
    atomicAdd(p + 1, v.y);
    atomicAdd(p + 2, v.z);
    atomicAdd(p + 3, v.w);
    if (c == 0) atomicAdd(&pcnt[g], 1.0f);
}

// ------------------------------------------------- tiny head: mean, MLP, relu
__global__ void k_head(const float* __restrict__ psum,
                       const float* __restrict__ pcnt,
                       const float* __restrict__ fw1, const float* __restrict__ fb1,
                       const float* __restrict__ fw2, const float* __restrict__ fb2,
                       float* __restrict__ out) {
    int g = threadIdx.x;
    if (g >= NUM_GRAPHS) return;
    const float inv = 1.0f / fmaxf(pcnt[g], 1.0f);
    float pl[HID];
    #pragma unroll
    for (int k = 0; k < HID; ++k) pl[k] = psum[(size_t)g * HID + k] * inv;
    float o = fb2[0];
    for (int j = 0; j < HID; ++j) {
        float hj = fb1[j];
        #pragma unroll
        for (int k = 0; k < HID; ++k) hj = fmaf(pl[k], fw1[k * HID + j], hj);
        o = fmaf(fmaxf(hj, 0.0f), fw2[j], o);
    }
    out[g] = fmaxf(o, 0.0f);
}

// ============================================================= launch
extern "C" void kernel_launch(void* const* d_in, const int* in_sizes, int n_in,
                              void* d_out, int out_size, void* d_ws, size_t ws_size,
                              hipStream_t stream) {
    (void)in_sizes; (void)n_in; (void)out_size; (void)ws_size;

    const float*     x     = (const float*)d_in[0];
    const long long* ei    = (const long long*)d_in[1];
    const long long* batch = (const long long*)d_in[2];
    const float* W1a[3] = {(const float*)d_in[3],  (const float*)d_in[9],  (const float*)d_in[15]};
    const float* B1a[3] = {(const float*)d_in[4],  (const float*)d_in[10], (const float*)d_in[16]};
    const float* W2a[3] = {(const float*)d_in[5],  (const float*)d_in[11], (const float*)d_in[17]};
    const float* B2a[3] = {(const float*)d_in[6],  (const float*)d_in[12], (const float*)d_in[18]};
    const float* Ga[3]  = {(const float*)d_in[7],  (const float*)d_in[13], (const float*)d_in[19]};
    const float* BTa[3] = {(const float*)d_in[8],  (const float*)d_in[14], (const float*)d_in[20]};
    const float* fw1 = (const float*)d_in[21];
    const float* fb1 = (const float*)d_in[22];
    const float* fw2 = (const float*)d_in[23];
    const float* fb2 = (const float*)d_in[24];

    float* ws    = (float*)d_ws;
    float* agg_y = ws;                                   // 6.4M floats (agg, then y in-place)
    float* hA    = ws + (size_t)N_NODES * HID;           // 6.4M floats
    float* stats = hA + (size_t)N_NODES * HID;           // 128 (sum, sumsq)
    float* bnp   = stats + 128;                          // 128 (scale, shift)
    float* psum  = bnp + 128;                            // 128*64
    float* pcnt  = psum + (size_t)NUM_GRAPHS * HID;      // 128

    const int elemGrid = (int)(((long long)N_NODES * 16 + 255) / 256);
    const int edgeGrid = (int)(((long long)N_EDGES * 16 + 255) / 256);

    const float* hin = x;
    for (int l = 0; l < 3; ++l) {
        k_zero<<<(N_NODES * HID + 255) / 256, 256, 0, stream>>>(agg_y, N_NODES * HID);
        k_zero<<<1, 128, 0, stream>>>(stats, 128);
        k_scatter<<<edgeGrid, 256, 0, stream>>>(hin, ei, agg_y);
        k_gin_mlp<<<(N_NODES + 63) / 64, 128, 0, stream>>>(
            hin, agg_y, W1a[l], B1a[l], W2a[l], B2a[l], agg_y /*y aliases agg*/, stats);
        k_bn_final<<<1, 64, 0, stream>>>(stats, Ga[l], BTa[l], bnp);
        k_bn_apply<<<elemGrid, 256, 0, stream>>>(agg_y, bnp, hA);
        hin = hA;
    }

    k_zero<<<(NUM_GRAPHS * HID + NUM_GRAPHS + 255) / 256, 256, 0, stream>>>(
        psum, NUM_GRAPHS * HID + NUM_GRAPHS);
    k_pool<<<elemGrid, 256, 0, stream>>>(hA, batch, psum, pcnt);
    k_head<<<1, 128, 0, stream>>>(psum, pcnt, fw1, fb1, fw2, fb2, (float*)d_out);
}